// ShareNN_25125558682013
// MI455X (gfx1250) — compile-verified
//
#include <hip/hip_runtime.h>
#include <hip/hip_bf16.h>

// Problem constants (match reference)
#define NMAT 384
#define DDIM 192
#define HDIM 192
#define VDIM 3
#define BDIM 16
#define ALPHA_C 15.0f
#define NSTEPS 20

typedef __attribute__((ext_vector_type(2))) float v2f;
typedef __attribute__((ext_vector_type(8))) float v8f;

static constexpr size_t NN   = (size_t)NMAT * NMAT;   // 147456
static constexpr size_t BNN  = (size_t)BDIM * NN;     // 2359296
static constexpr size_t VBNN = (size_t)VDIM * BNN;    // 7077888
static constexpr size_t ND   = (size_t)NMAT * DDIM;   // 73728

__device__ __forceinline__ v8f wmma4(v2f a, v2f b, v8f c) {
  return __builtin_amdgcn_wmma_f32_16x16x4_f32(false, a, false, b, (short)0, c,
                                               false, false);
}

// ---------------------------------------------------------------------------
// Core: one wave computes a 32x32 macro-tile of C = op(A) @ op(B) as a 2x2
// grid of 16x16 V_WMMA_F32_16X16X4_F32 accumulators. A-fragments are reused
// across two column tiles and B-fragments across two row tiles (2x operand
// reuse vs a single-tile wave). Fragment layouts per CDNA5 ISA 7.12.2.
// ---------------------------------------------------------------------------
template <int TRANSA, int TRANSB>
__device__ __forceinline__ void wmma_tile32(const float* __restrict__ A, int lda,
                                            const float* __restrict__ B, int ldb,
                                            int K, int tm, int tn,
                                            v8f& c00, v8f& c01, v8f& c10, v8f& c11) {
  const int lane = threadIdx.x & 31;
  const int half = lane >> 4;   // 0 or 1
  const int l15  = lane & 15;
  const int r0 = tm + l15, r1 = tm + 16 + l15;
  const int cc0 = tn + l15, cc1 = tn + 16 + l15;
  c00 = v8f{}; c01 = v8f{}; c10 = v8f{}; c11 = v8f{};
  for (int k = 0; k < K; k += 4) {
    const int ka = k + half * 2;
    v2f a0, a1, b0, b1;
    if (TRANSA) {
      a0.x = A[(size_t)(ka + 0) * lda + r0]; a0.y = A[(size_t)(ka + 1) * lda + r0];
      a1.x = A[(size_t)(ka + 0) * lda + r1]; a1.y = A[(size_t)(ka + 1) * lda + r1];
    } else {
      a0.x = A[(size_t)r0 * lda + ka]; a0.y = A[(size_t)r0 * lda + ka + 1];
      a1.x = A[(size_t)r1 * lda + ka]; a1.y = A[(size_t)r1 * lda + ka + 1];
    }
    if (TRANSB) {
      b0.x = B[(size_t)cc0 * ldb + ka]; b0.y = B[(size_t)cc0 * ldb + ka + 1];
      b1.x = B[(size_t)cc1 * ldb + ka]; b1.y = B[(size_t)cc1 * ldb + ka + 1];
    } else {
      b0.x = B[(size_t)(ka + 0) * ldb + cc0]; b0.y = B[(size_t)(ka + 1) * ldb + cc0];
      b1.x = B[(size_t)(ka + 0) * ldb + cc1]; b1.y = B[(size_t)(ka + 1) * ldb + cc1];
    }
    c00 = wmma4(a0, b0, c00);
    c01 = wmma4(a0, b1, c01);
    c10 = wmma4(a1, b0, c10);
    c11 = wmma4(a1, b1, c11);
  }
}

// ---------------------------------------------------------------------------
// a_ls[vb] = adjs[vb] + adjs[vb] @ adjs[vb]        (48 batches, 384^3)
// ---------------------------------------------------------------------------
__global__ __launch_bounds__(256) void k_a_ls(const float* __restrict__ adjs,
                                              float* __restrict__ a_ls) {
  const int wid = (blockIdx.x << 3) + (threadIdx.x >> 5);
  const int tilesPer = 12 * 12;
  const int batch = wid / tilesPer;
  const int t = wid % tilesPer;
  const int tm = (t / 12) * 32, tn = (t % 12) * 32;
  const float* A = adjs + (size_t)batch * NN;
  v8f c00, c01, c10, c11;
  wmma_tile32<0, 0>(A, NMAT, A, NMAT, NMAT, tm, tn, c00, c01, c10, c11);
  const int lane = threadIdx.x & 31;
  const int half = lane >> 4, l15 = lane & 15;
  float* C = a_ls + (size_t)batch * NN;
  const int cc0 = tn + l15, cc1 = tn + 16 + l15;
#pragma unroll
  for (int i = 0; i < 8; ++i) {
    const int row0 = tm + half * 8 + i;
    const int row1 = row0 + 16;
    C[(size_t)row0 * NMAT + cc0] = c00[i] + A[(size_t)row0 * NMAT + cc0];
    C[(size_t)row0 * NMAT + cc1] = c01[i] + A[(size_t)row0 * NMAT + cc1];
    C[(size_t)row1 * NMAT + cc0] = c10[i] + A[(size_t)row1 * NMAT + cc0];
    C[(size_t)row1 * NMAT + cc1] = c11[i] + A[(size_t)row1 * NMAT + cc1];
  }
}

// ---------------------------------------------------------------------------
// xxt_v[vb] = X @ X^T where X = xs[vb] (384x192)   (48 batches)
// ---------------------------------------------------------------------------
__global__ __launch_bounds__(256) void k_xxt(const float* __restrict__ xs,
                                             float* __restrict__ xxt) {
  const int wid = (blockIdx.x << 3) + (threadIdx.x >> 5);
  const int tilesPer = 12 * 12;
  const int batch = wid / tilesPer;
  const int t = wid % tilesPer;
  const int tm = (t / 12) * 32, tn = (t % 12) * 32;
  const float* X = xs + (size_t)batch * ND;
  v8f c00, c01, c10, c11;
  wmma_tile32<0, 1>(X, DDIM, X, DDIM, DDIM, tm, tn, c00, c01, c10, c11);
  const int lane = threadIdx.x & 31;
  const int half = lane >> 4, l15 = lane & 15;
  float* C = xxt + (size_t)batch * NN;
  const int cc0 = tn + l15, cc1 = tn + 16 + l15;
#pragma unroll
  for (int i = 0; i < 8; ++i) {
    const int row0 = tm + half * 8 + i;
    const int row1 = row0 + 16;
    C[(size_t)row0 * NMAT + cc0] = c00[i];
    C[(size_t)row0 * NMAT + cc1] = c01[i];
    C[(size_t)row1 * NMAT + cc0] = c10[i];
    C[(size_t)row1 * NMAT + cc1] = c11[i];
  }
}

// ---------------------------------------------------------------------------
// M[b]   = alpha*Isum*I + sum_v lam[v]*xxt_v[v,b]
// RHS[b] = alpha*sum_v lam[v]*a_ls[v,b] + sum_v lam[v]*xxt_v[v,b]
// lam layout: lam[0..2] = lambdas, lam[3] = Isum
// ---------------------------------------------------------------------------
__global__ __launch_bounds__(256) void k_prep(const float* __restrict__ xxt,
                                              const float* __restrict__ a_ls,
                                              const float* __restrict__ lam,
                                              float* __restrict__ Mm,
                                              float* __restrict__ RHS) {
  const size_t idx = (size_t)blockIdx.x * 256 + threadIdx.x;  // over B*N*N
  const float l0 = lam[0], l1 = lam[1], l2 = lam[2], isum = lam[3];
  const float xx = l0 * xxt[idx] + l1 * xxt[idx + BNN] + l2 * xxt[idx + 2 * BNN];
  const float fa = l0 * a_ls[idx] + l1 * a_ls[idx + BNN] + l2 * a_ls[idx + 2 * BNN];
  const size_t nm = idx % NN;
  const int n = (int)(nm / NMAT), m = (int)(nm % NMAT);
  Mm[idx]  = xx + ((n == m) ? ALPHA_C * isum : 0.0f);
  RHS[idx] = ALPHA_C * fa + xx;
}

// ---------------------------------------------------------------------------
// In-place Gauss-Jordan inverse of an SPD 384x384 matrix; one WG per matrix.
// 768 threads = (row, column-half); pivot row AND pivot column staged in LDS
// in one phase, so each pivot needs only 2 barriers; row updates are 48
// float4 FMAs per thread.
// ---------------------------------------------------------------------------
__global__ __launch_bounds__(768) void k_invert(float* __restrict__ Mm) {
  __shared__ __align__(16) float rowk[NMAT];
  __shared__ float colk[NMAT];
  float* A = Mm + (size_t)blockIdx.x * NN;
  const int tid = threadIdx.x;   // 0..767
  const int i   = tid >> 1;      // row 0..383
  const int hlf = tid & 1;       // column half: cols [hlf*192, hlf*192+192)
  for (int k = 0; k < NMAT; ++k) {
    if (tid < NMAT) rowk[tid] = A[(size_t)k * NMAT + tid];
    else            colk[tid - NMAT] = A[(size_t)(tid - NMAT) * NMAT + k];
    __syncthreads();
    const float ip = 1.0f / rowk[k];
    // scale pivot row (A[k][k] becomes 1/pivot)
    if (tid < NMAT) A[(size_t)k * NMAT + tid] = (tid == k) ? ip : rowk[tid] * ip;
    // eliminate all other rows; each (row, half) owned by exactly one thread
    if (i != k) {
      const float fip = colk[i] * ip;
      float4* Ar = (float4*)(A + (size_t)i * NMAT) + hlf * 48;
      const float4* Rk = (const float4*)rowk + hlf * 48;
#pragma unroll 4
      for (int j4 = 0; j4 < 48; ++j4) {
        float4 a = Ar[j4];
        const float4 r = Rk[j4];
        a.x -= fip * r.x; a.y -= fip * r.y; a.z -= fip * r.z; a.w -= fip * r.w;
        Ar[j4] = a;
      }
      if (((k >= 192) ? 1 : 0) == hlf) A[(size_t)i * NMAT + k] = -fip;  // col-k fixup
    }
    __syncthreads();
  }
}

// ---------------------------------------------------------------------------
// S[b] = Minv[b] @ RHS[b]        (16 batches, 384^3)
// ---------------------------------------------------------------------------
__global__ __launch_bounds__(256) void k_S(const float* __restrict__ Minv,
                                           const float* __restrict__ RHS,
                                           float* __restrict__ S) {
  const int wid = (blockIdx.x << 3) + (threadIdx.x >> 5);
  const int tilesPer = 12 * 12;
  const int batch = wid / tilesPer;
  const int t = wid % tilesPer;
  const int tm = (t / 12) * 32, tn = (t % 12) * 32;
  const float* A = Minv + (size_t)batch * NN;
  const float* B = RHS + (size_t)batch * NN;
  v8f c00, c01, c10, c11;
  wmma_tile32<0, 0>(A, NMAT, B, NMAT, NMAT, tm, tn, c00, c01, c10, c11);
  const int lane = threadIdx.x & 31;
  const int half = lane >> 4, l15 = lane & 15;
  float* C = S + (size_t)batch * NN;
  const int cc0 = tn + l15, cc1 = tn + 16 + l15;
#pragma unroll
  for (int i = 0; i < 8; ++i) {
    const int row0 = tm + half * 8 + i;
    const int row1 = row0 + 16;
    C[(size_t)row0 * NMAT + cc0] = c00[i];
    C[(size_t)row0 * NMAT + cc1] = c01[i];
    C[(size_t)row1 * NMAT + cc0] = c10[i];
    C[(size_t)row1 * NMAT + cc1] = c11[i];
  }
}

// ---------------------------------------------------------------------------
// e1: recon = xs[v,b] - S[v]^T @ xs[v,b]; accumulate sum(recon^2) into e[v].
// GEMM is (S^T)(384x384) @ X(384x192); A accessed transposed.
// ---------------------------------------------------------------------------
__global__ __launch_bounds__(256) void k_e1(const float* __restrict__ S,
                                            const float* __restrict__ xs,
                                            float* __restrict__ e) {
  const int wid = (blockIdx.x << 3) + (threadIdx.x >> 5);
  const int tilesPer = 12 * 6;
  const int vb = wid / tilesPer;
  const int t = wid % tilesPer;
  const int tm = (t / 6) * 32, tn = (t % 6) * 32;
  const int v = vb / BDIM;
  const float* A = S + (size_t)v * NN;          // transposed access
  const float* X = xs + (size_t)vb * ND;
  v8f c00, c01, c10, c11;
  wmma_tile32<1, 0>(A, NMAT, X, DDIM, NMAT, tm, tn, c00, c01, c10, c11);
  const int lane = threadIdx.x & 31;
  const int half = lane >> 4, l15 = lane & 15;
  const int cc0 = tn + l15, cc1 = tn + 16 + l15;
  float s = 0.0f;
#pragma unroll
  for (int i = 0; i < 8; ++i) {
    const int row0 = tm + half * 8 + i;
    const int row1 = row0 + 16;
    float r;
    r = X[(size_t)row0 * DDIM + cc0] - c00[i]; s += r * r;
    r = X[(size_t)row0 * DDIM + cc1] - c01[i]; s += r * r;
    r = X[(size_t)row1 * DDIM + cc0] - c10[i]; s += r * r;
    r = X[(size_t)row1 * DDIM + cc1] - c11[i]; s += r * r;
  }
  for (int off = 16; off > 0; off >>= 1) s += __shfl_xor(s, off, 32);
  if (lane == 0) atomicAdd(&e[v], s);
}

// ---------------------------------------------------------------------------
// e2: e[v] += alpha * sum_{b,n,m} (S[v][n,m] - a_ls[v,b][n,m])^2
// grid: (BNN/256, V)
// ---------------------------------------------------------------------------
__global__ __launch_bounds__(256) void k_e2(const float* __restrict__ S,
                                            const float* __restrict__ a_ls,
                                            float* __restrict__ e) {
  const int v = blockIdx.y;
  const size_t idx = (size_t)blockIdx.x * 256 + threadIdx.x;  // over B*N*N
  const size_t nm = idx % NN;
  const float d = S[(size_t)v * NN + nm] - a_ls[(size_t)v * BNN + idx];
  float p = ALPHA_C * d * d;
  __shared__ float sd[256];
  const int tid = threadIdx.x;
  sd[tid] = p;
  __syncthreads();
  for (int s = 128; s > 0; s >>= 1) {
    if (tid < s) sd[tid] += sd[tid + s];
    __syncthreads();
  }
  if (tid == 0) atomicAdd(&e[v], sd[0]);
}

// ---------------------------------------------------------------------------
// lambda update: gamma = -1  =>  lam[v] = e[v]^(-1/2). Also Isum; reset e.
// ---------------------------------------------------------------------------
__global__ void k_lambda(float* __restrict__ lam, float* __restrict__ e) {
  if (threadIdx.x == 0) {
    float s = 0.0f;
#pragma unroll
    for (int v = 0; v < VDIM; ++v) {
      const float l = rsqrtf(e[v]);
      lam[v] = l;
      s += l;
      e[v] = 0.0f;
    }
    lam[3] = s;
  }
}

__global__ void k_init(float* __restrict__ lam, float* __restrict__ e) {
  if (threadIdx.x == 0) {
    lam[0] = 1.0f; lam[1] = 1.0f; lam[2] = 1.0f; lam[3] = (float)VDIM;
    e[0] = 0.0f; e[1] = 0.0f; e[2] = 0.0f;
  }
}

// ---------------------------------------------------------------------------
// out[b] = S[b] @ W + bias       (16 batches, 384x384 @ 384x192)
// ---------------------------------------------------------------------------
__global__ __launch_bounds__(256) void k_out(const float* __restrict__ S,
                                             const float* __restrict__ W,
                                             const float* __restrict__ bias,
                                             float* __restrict__ out) {
  const int wid = (blockIdx.x << 3) + (threadIdx.x >> 5);
  const int tilesPer = 12 * 6;
  const int batch = wid / tilesPer;
  const int t = wid % tilesPer;
  const int tm = (t / 6) * 32, tn = (t % 6) * 32;
  const float* A = S + (size_t)batch * NN;
  v8f c00, c01, c10, c11;
  wmma_tile32<0, 0>(A, NMAT, W, HDIM, NMAT, tm, tn, c00, c01, c10, c11);
  const int lane = threadIdx.x & 31;
  const int half = lane >> 4, l15 = lane & 15;
  const int cc0 = tn + l15, cc1 = tn + 16 + l15;
  float* C = out + (size_t)batch * NMAT * HDIM;
  const float bv0 = bias[cc0], bv1 = bias[cc1];
#pragma unroll
  for (int i = 0; i < 8; ++i) {
    const int row0 = tm + half * 8 + i;
    const int row1 = row0 + 16;
    C[(size_t)row0 * HDIM + cc0] = c00[i] + bv0;
    C[(size_t)row0 * HDIM + cc1] = c01[i] + bv1;
    C[(size_t)row1 * HDIM + cc0] = c10[i] + bv0;
    C[(size_t)row1 * HDIM + cc1] = c11[i] + bv1;
  }
}

// ---------------------------------------------------------------------------
extern "C" void kernel_launch(void* const* d_in, const int* in_sizes, int n_in,
                              void* d_out, int out_size, void* d_ws, size_t ws_size,
                              hipStream_t stream) {
  (void)in_sizes; (void)n_in; (void)out_size; (void)ws_size;
  const float* xs   = (const float*)d_in[0];  // (V,B,N,D)
  const float* adjs = (const float*)d_in[1];  // (V,B,N,N)
  const float* W    = (const float*)d_in[2];  // (N,H)
  const float* bias = (const float*)d_in[3];  // (H)
  float* out = (float*)d_out;                 // (B,N,H)

  float* ws   = (float*)d_ws;
  float* a_ls = ws;                 // VBNN
  float* xxt  = ws + VBNN;          // VBNN
  float* Mm   = ws + 2 * VBNN;      // BNN
  float* RHS  = Mm + BNN;           // BNN
  float* S    = RHS + BNN;          // BNN
  float* lam  = S + BNN;            // 4 floats (lam0..2, Isum)
  float* e    = lam + 8;            // 3 floats

  k_init<<<1, 32, 0, stream>>>(lam, e);
  k_a_ls<<<(48 * 144) / 8, 256, 0, stream>>>(adjs, a_ls);
  k_xxt <<<(48 * 144) / 8, 256, 0, stream>>>(xs, xxt);

  for (int s = 0; s < NSTEPS; ++s) {
    k_prep  <<<(int)(BNN / 256), 256, 0, stream>>>(xxt, a_ls, lam, Mm, RHS);
    k_invert<<<BDIM, 768, 0, stream>>>(Mm);
    k_S     <<<(16 * 144) / 8, 256, 0, stream>>>(Mm, RHS, S);
    if (s + 1 < NSTEPS) {  // last step's lambdas are unused by the output
      k_e1    <<<(48 * 72) / 8, 256, 0, stream>>>(S, xs, e);
      k_e2    <<<dim3((unsigned)(BNN / 256), VDIM), 256, 0, stream>>>(S, a_ls, e);
      k_lambda<<<1, 32, 0, stream>>>(lam, e);
    }
  }
  k_out<<<(16 * 72) / 8, 256, 0, stream>>>(S, W, bias, out);
}